// MaskedClassifiers_18090402250892
// MI455X (gfx1250) — compile-verified
//
#include <hip/hip_runtime.h>

// MI455X / gfx1250, wave32, WMMA (not MFMA).
typedef float v2f __attribute__((ext_vector_type(2)));
typedef float v8f __attribute__((ext_vector_type(8)));

constexpr int N_ROWS = 131072;
constexpr int DIM    = 512;
constexpr int CTOT   = 43;        // 17 + 12 + 6 + 8
constexpr int CPAD   = 48;        // 3 tiles of 16 columns
constexpr int KPAIRS = DIM / 2;   // 256 pairs of K values
constexpr int WAVES_PER_BLOCK = 8;
constexpr int ROWS_PER_WAVE   = 16;
constexpr int ROWS_PER_BLOCK  = WAVES_PER_BLOCK * ROWS_PER_WAVE; // 128
constexpr int GRID            = N_ROWS / ROWS_PER_BLOCK;         // 1024
constexpr size_t LDS_BYTES    = (size_t)KPAIRS * CPAD * sizeof(v2f); // 96 KB

__device__ __forceinline__ void col_to_head(int c, int& h, int& lc) {
  if (c < 17)      { h = 0; lc = c;      }
  else if (c < 29) { h = 1; lc = c - 17; }
  else if (c < 35) { h = 2; lc = c - 29; }
  else             { h = 3; lc = c - 35; }
}

__global__ __launch_bounds__(256) void masked_heads_wmma_f32(
    const float* __restrict__ x,  const int* __restrict__ mask,
    const float* __restrict__ W0, const float* __restrict__ b0,
    const float* __restrict__ W1, const float* __restrict__ b1,
    const float* __restrict__ W2, const float* __restrict__ b2,
    const float* __restrict__ W3, const float* __restrict__ b3,
    float* __restrict__ out)
{
  // LDS B tile in WMMA-fragment-friendly layout:
  // ldsb[kpair * CPAD + col] = (W[2*kpair][col], W[2*kpair+1][col])
  extern __shared__ v2f ldsb[];

  const int tid = threadIdx.x;

  // ---- Pack concatenated/padded weights into LDS (uniform trip count) ----
  const float* Ws[4] = {W0, W1, W2, W3};
  const int    Cs[4] = {17, 12, 6, 8};
  for (int idx = tid; idx < KPAIRS * CPAD; idx += 256) {
    const int kp = idx / CPAD;
    const int c  = idx % CPAD;
    float w0 = 0.0f, w1 = 0.0f;
    if (c < CTOT) {
      int h, lc; col_to_head(c, h, lc);
      const float* W = Ws[h];
      const int    ci = Cs[h];
      w0 = W[(size_t)(2 * kp + 0) * ci + lc];
      w1 = W[(size_t)(2 * kp + 1) * ci + lc];
    }
    v2f p; p.x = w0; p.y = w1;
    ldsb[idx] = p;
  }
  __syncthreads();

  // ---- Per-wave 16-row tile ----
  const int lane = tid & 31;          // wave32
  const int wave = tid >> 5;
  const int l15  = lane & 15;
  const int hi   = lane >> 4;         // lane half selects K sub-pair (A) / K rows (B)
  const int row_base = (blockIdx.x * WAVES_PER_BLOCK + wave) * ROWS_PER_WAVE;

  // A fragment source: lane l<16 supplies rows M=l, K=4g+{0,1};
  // lanes 16-31 supply rows M=l-16, K=4g+{2,3}.  One b64 load per K-group.
  // x is single-use streaming data -> non-temporal loads (don't churn L2).
  const v2f* __restrict__ ap =
      (const v2f*)(x + (size_t)(row_base + l15) * DIM + 2 * hi);

  v8f acc0 = {0.f,0.f,0.f,0.f,0.f,0.f,0.f,0.f};
  v8f acc1 = {0.f,0.f,0.f,0.f,0.f,0.f,0.f,0.f};
  v8f acc2 = {0.f,0.f,0.f,0.f,0.f,0.f,0.f,0.f};

  // K loop: 128 groups of K=4, 3 column tiles per group.
#pragma unroll 8
  for (int g = 0; g < DIM / 4; ++g) {
    const v2f a = __builtin_nontemporal_load(ap + 2 * g);
    const v2f* __restrict__ bg = ldsb + (size_t)(2 * g + hi) * CPAD + l15;
    const v2f bA = bg[0];
    const v2f bB = bg[16];
    const v2f bC = bg[32];
    acc0 = __builtin_amdgcn_wmma_f32_16x16x4_f32(false, a, false, bA,
                                                 (short)0, acc0, false, false);
    acc1 = __builtin_amdgcn_wmma_f32_16x16x4_f32(false, a, false, bB,
                                                 (short)0, acc1, false, false);
    acc2 = __builtin_amdgcn_wmma_f32_16x16x4_f32(false, a, false, bC,
                                                 (short)0, acc2, false, false);
  }

  // ---- Store with bias + mask-routing, drop padded columns ----
  // C/D layout: lane l, VGPR r -> row = r + 8*(l>=16), col = 16*t + (l&15).
  const int row0 = row_base + hi * 8;
  int m[8];
#pragma unroll
  for (int r = 0; r < 8; ++r) m[r] = mask[row0 + r];

  const float* bs[4] = {b0, b1, b2, b3};
  v8f accs[3] = {acc0, acc1, acc2};

#pragma unroll
  for (int t = 0; t < 3; ++t) {
    const int c = t * 16 + l15;
    if (c < CTOT) {
      int h, lc; col_to_head(c, h, lc);
      const float bias = bs[h][lc];
#pragma unroll
      for (int r = 0; r < 8; ++r) {
        const int   row = row0 + r;
        const float v   = accs[t][r] + bias;
        const float o   = (m[r] == h) ? v : 0.0f;
        // Output is write-once streaming data -> non-temporal store.
        __builtin_nontemporal_store(o, &out[(size_t)row * CTOT + c]);
      }
    }
  }
}

extern "C" void kernel_launch(void* const* d_in, const int* in_sizes, int n_in,
                              void* d_out, int out_size, void* d_ws, size_t ws_size,
                              hipStream_t stream) {
  (void)in_sizes; (void)n_in; (void)d_ws; (void)ws_size; (void)out_size;

  // setup_inputs() dict order: x, mask, W0, b0, W1, b1, W2, b2, W3, b3
  const float* x    = (const float*)d_in[0];
  const int*   mask = (const int*)  d_in[1];
  const float* W0   = (const float*)d_in[2];
  const float* b0   = (const float*)d_in[3];
  const float* W1   = (const float*)d_in[4];
  const float* b1   = (const float*)d_in[5];
  const float* W2   = (const float*)d_in[6];
  const float* b2   = (const float*)d_in[7];
  const float* W3   = (const float*)d_in[8];
  const float* b3   = (const float*)d_in[9];
  float*       out  = (float*)d_out;

  // Allow the 96 KB dynamic-LDS allocation (WGP has 320 KB; 3 blocks/WGP).
  (void)hipFuncSetAttribute((const void*)masked_heads_wmma_f32,
                            hipFuncAttributeMaxDynamicSharedMemorySize,
                            (int)LDS_BYTES);

  masked_heads_wmma_f32<<<dim3(GRID), dim3(256), LDS_BYTES, stream>>>(
      x, mask, W0, b0, W1, b1, W2, b2, W3, b3, out);
}